// QuantumRegressionModel_65481071403893
// MI455X (gfx1250) — compile-verified
//
#include <hip/hip_runtime.h>
#include <hip/hip_bf16.h>

// MI455X / gfx1250, wave32. Quantum circuit features + linear head.
// - 4 adjacent single-qubit gates fuse into a 16x16 complex unitary (kron),
//   applied with V_WMMA_F32_16X16X4_F32 (4 K-chunks x 4 real products).
// - Two unitary groups fused per kernel via a 33KB LDS tile (phase A: global
//   -> WMMA -> LDS along the high nibble; phase B: LDS -> WMMA -> global
//   along the low nibble). 6 state passes total; state (64MB) lives in L2.
// - CNOT ring == GF(2)-linear index permutation, fused into next pass reads.

typedef float v2f __attribute__((ext_vector_type(2)));
typedef float v8f __attribute__((ext_vector_type(8)));

#define NW    16
#define QDIM  65536
#define BATCH 128
#define HSTR  260          // padded h-stride (floats): 8*260 % 64 == 32 -> no bank conflicts
#define LDSZ  (15*HSTR + 15*16 + 15 + 1)

// CNOT-ring gather map: psi_after[j] = psi_before[perm_src(j)]
__device__ __forceinline__ unsigned perm_src(unsigned j) {
    unsigned g = j ^ (j >> 1);
    return g ^ ((j & 1u) ? 0xC000u : 0u);
}

// ---------------------------------------------------------------------------
// Build 12 fused 16x16 complex unitaries: U[d][g] = kron of 4 RZ*RY gates.
__global__ void build_unitaries_kernel(const float* __restrict__ params,
                                       float* __restrict__ Ure,
                                       float* __restrict__ Uim) {
    int mat = blockIdx.x;            // d*4 + g
    int d = mat >> 2, g = mat & 3;
    int e = threadIdx.x;
    int i = e >> 4, j = e & 15;      // row (new), col (old) of 4-bit group axis
    float ar = 1.f, ai = 0.f;
    #pragma unroll
    for (int q = 0; q < 4; ++q) {    // wire 4g+q <-> bit (3-q) of group axis
        int w = 4 * g + q;
        float th = 0.5f * params[(d * NW + w) * 2 + 0];
        float ph = 0.5f * params[(d * NW + w) * 2 + 1];
        float ct = cosf(th), st = sinf(th);
        float er = cosf(ph), ei = sinf(ph);   // em = er - i*ei, ep = er + i*ei
        int bi = (i >> (3 - q)) & 1;
        int bj = (j >> (3 - q)) & 1;
        float mr, mi;
        if (!bi) { float m = bj ? -st : ct; mr = er * m; mi = -ei * m; }  // em*{ct,-st}
        else     { float m = bj ?  ct : st; mr = er * m; mi =  ei * m; }  // ep*{st, ct}
        float nr = ar * mr - ai * mi;
        float ni = ar * mi + ai * mr;
        ar = nr; ai = ni;
    }
    Ure[mat * 256 + i * 16 + j] = ar;
    Uim[mat * 256 + i * 16 + j] = ai;
}

// ---------------------------------------------------------------------------
// Element index for (hi-nibble value k, lo-nibble value l, remaining-8-bit o8).
template <bool HI>
__device__ __forceinline__ unsigned pair_idx(unsigned k, unsigned l, unsigned o8) {
    return HI ? ((k << 12) | (l << 8) | o8)        // nibbles at bits 15-12 / 11-8
              : ((o8 << 8) | (k << 4) | l);        // nibbles at bits  7-4 /  3-0
}

// Load A fragments (16x4 f32 per chunk) of a 16x16 unitary, plus negated Im.
__device__ __forceinline__ void load_afrag(const float* __restrict__ Ure,
                                           const float* __restrict__ Uim,
                                           unsigned n, unsigned kb,
                                           v2f aRe[4], v2f aIm[4], v2f aImN[4]) {
    #pragma unroll
    for (int c = 0; c < 4; ++c) {
        unsigned k0 = 4 * c + kb;
        aRe[c][0] = Ure[n * 16 + k0];  aRe[c][1] = Ure[n * 16 + k0 + 1];
        aIm[c][0] = Uim[n * 16 + k0];  aIm[c][1] = Uim[n * 16 + k0 + 1];
        aImN[c][0] = -aIm[c][0];       aImN[c][1] = -aIm[c][1];
    }
}

// ---------------------------------------------------------------------------
// Apply two 16x16 complex unitaries (hi nibble then lo nibble) in one pass.
// Block = 256 threads (8 wave32) owns a self-contained 256x16-column complex
// tile: phase A global->LDS (transform along hi nibble), barrier, phase B
// LDS->global (transform along lo nibble). In-place safe when !PERM.
template <bool IM_ZERO, bool PERM, bool HI>
__launch_bounds__(256)
__global__ void apply_pair_kernel(const float* __restrict__ src_re,
                                  const float* __restrict__ src_im,
                                  float* __restrict__ dst_re,
                                  float* __restrict__ dst_im,
                                  const float* __restrict__ Uhi_re,
                                  const float* __restrict__ Uhi_im,
                                  const float* __restrict__ Ulo_re,
                                  const float* __restrict__ Ulo_im) {
    __shared__ float Lre[LDSZ];
    __shared__ float Lim[LDSZ];

    const unsigned lane = threadIdx.x & 31;
    const unsigned wv   = threadIdx.x >> 5;
    const unsigned n    = lane & 15;           // A row / B,D column
    const unsigned kb   = (lane >> 4) << 1;    // K offset from lane half
    const unsigned rowadd = (lane >> 4) << 3;  // D rows r / r+8
    const unsigned flat = blockIdx.x;          // 2048 = 128 batch x 16 col-groups
    const unsigned b    = flat >> 4;
    const unsigned t    = flat & 15;
    const unsigned o8   = t * 16u + n;         // remaining 8-bit index
    const unsigned base = b * (unsigned)QDIM;

    v2f aRe[4], aIm[4], aImN[4];

    // ---- phase A: transform along hi nibble, global -> LDS ----
    load_afrag(Uhi_re, Uhi_im, n, kb, aRe, aIm, aImN);
    #pragma unroll
    for (int tt = 0; tt < 2; ++tt) {
        unsigned l = 2 * wv + (unsigned)tt;    // fixed lo-nibble value of this tile
        v8f dRe = {0.f,0.f,0.f,0.f,0.f,0.f,0.f,0.f};
        v8f dIm = {0.f,0.f,0.f,0.f,0.f,0.f,0.f,0.f};
        #pragma unroll
        for (int c = 0; c < 4; ++c) {
            unsigned k0 = 4 * c + kb;
            unsigned i0 = pair_idx<HI>(k0,     l, o8);
            unsigned i1 = pair_idx<HI>(k0 + 1, l, o8);
            if (PERM) { i0 = perm_src(i0); i1 = perm_src(i1); }
            v2f bRe, bIm;
            bRe[0] = src_re[base + i0];
            bRe[1] = src_re[base + i1];
            dRe = __builtin_amdgcn_wmma_f32_16x16x4_f32(false, aRe[c], false, bRe,
                                                        (short)0, dRe, false, false);
            dIm = __builtin_amdgcn_wmma_f32_16x16x4_f32(false, aIm[c], false, bRe,
                                                        (short)0, dIm, false, false);
            if (!IM_ZERO) {
                bIm[0] = src_im[base + i0];
                bIm[1] = src_im[base + i1];
                dRe = __builtin_amdgcn_wmma_f32_16x16x4_f32(false, aImN[c], false, bIm,
                                                            (short)0, dRe, false, false);
                dIm = __builtin_amdgcn_wmma_f32_16x16x4_f32(false, aRe[c], false, bIm,
                                                            (short)0, dIm, false, false);
            }
        }
        #pragma unroll
        for (int r = 0; r < 8; ++r) {
            unsigned h  = (unsigned)r + rowadd;
            unsigned ad = h * HSTR + l * 16u + n;
            Lre[ad] = dRe[r];
            Lim[ad] = dIm[r];
        }
    }
    __syncthreads();

    // ---- phase B: transform along lo nibble, LDS -> global ----
    load_afrag(Ulo_re, Ulo_im, n, kb, aRe, aIm, aImN);
    #pragma unroll
    for (int tt = 0; tt < 2; ++tt) {
        unsigned h = 2 * wv + (unsigned)tt;    // fixed hi-nibble value of this tile
        v8f dRe = {0.f,0.f,0.f,0.f,0.f,0.f,0.f,0.f};
        v8f dIm = {0.f,0.f,0.f,0.f,0.f,0.f,0.f,0.f};
        #pragma unroll
        for (int c = 0; c < 4; ++c) {
            unsigned k0 = 4 * c + kb;
            unsigned a0 = h * HSTR + k0 * 16u + n;
            unsigned a1 = a0 + 16u;
            v2f bRe, bIm;
            bRe[0] = Lre[a0];  bRe[1] = Lre[a1];
            bIm[0] = Lim[a0];  bIm[1] = Lim[a1];
            dRe = __builtin_amdgcn_wmma_f32_16x16x4_f32(false, aRe[c], false, bRe,
                                                        (short)0, dRe, false, false);
            dIm = __builtin_amdgcn_wmma_f32_16x16x4_f32(false, aIm[c], false, bRe,
                                                        (short)0, dIm, false, false);
            dRe = __builtin_amdgcn_wmma_f32_16x16x4_f32(false, aImN[c], false, bIm,
                                                        (short)0, dRe, false, false);
            dIm = __builtin_amdgcn_wmma_f32_16x16x4_f32(false, aRe[c], false, bIm,
                                                        (short)0, dIm, false, false);
        }
        #pragma unroll
        for (int r = 0; r < 8; ++r) {
            unsigned lo  = (unsigned)r + rowadd;
            unsigned idx = pair_idx<HI>(h, lo, o8);
            dst_re[base + idx] = dRe[r];
            dst_im[base + idx] = dIm[r];
        }
    }
}

// ---------------------------------------------------------------------------
// Final: fuse last CNOT-ring perm (forward map via prefix-XOR fold) + probs +
// PauliZ signs + linear head. One block per batch element.
__global__ void features_kernel(const float* __restrict__ psi_re,
                                const float* __restrict__ psi_im,
                                const float* __restrict__ head_w,
                                const float* __restrict__ head_b,
                                float* __restrict__ out) {
    __shared__ float hw[16];
    __shared__ float red[256];
    int b = blockIdx.x;
    int tid = threadIdx.x;
    if (tid < 16) hw[tid] = head_w[tid];
    __syncthreads();
    const float* pr = psi_re + (size_t)b * QDIM;
    const float* pi = psi_im + (size_t)b * QDIM;
    float acc = 0.f;
    for (int it = 0; it < QDIM / 256; ++it) {
        unsigned j = (unsigned)(it * 256 + tid);
        float re = pr[j], im = pi[j];
        float p = re * re + im * im;
        // forward CNOT-ring map y = F(j): bit p (p<=14) = XOR of j bits p..15,
        // bit 15 = XOR of j bits 0..14
        unsigned pf = j;
        pf ^= pf >> 1; pf ^= pf >> 2; pf ^= pf >> 4; pf ^= pf >> 8;
        unsigned y = (pf & 0x7FFFu) | (((pf ^ (j >> 15)) & 1u) << 15);
        float S = 0.f;
        #pragma unroll
        for (int w = 0; w < 16; ++w)
            S += ((y >> (15 - w)) & 1u) ? -hw[w] : hw[w];
        acc += p * S;
    }
    red[tid] = acc;
    __syncthreads();
    #pragma unroll
    for (int s = 128; s > 0; s >>= 1) {
        if (tid < s) red[tid] += red[tid + s];
        __syncthreads();
    }
    if (tid == 0) out[b] = red[0] + head_b[0];
}

// ---------------------------------------------------------------------------
extern "C" void kernel_launch(void* const* d_in, const int* in_sizes, int n_in,
                              void* d_out, int out_size, void* d_ws, size_t ws_size,
                              hipStream_t stream) {
    const float* state  = (const float*)d_in[0];   // (128, 65536) f32
    const float* params = (const float*)d_in[1];   // (3, 16, 2)   f32
    const float* head_w = (const float*)d_in[2];   // (1, 16)      f32
    const float* head_b = (const float*)d_in[3];   // (1,)         f32
    float* out = (float*)d_out;                    // (128,)       f32

    // workspace (~128.03 MB): 12 unitaries + A/B ping-pong complex state
    char* ws = (char*)d_ws;
    float* Ure = (float*)ws;                 // 12*256 f32
    float* Uim = Ure + 12 * 256;             // 12*256 f32
    size_t off = 32768;
    const size_t plane = (size_t)BATCH * QDIM * sizeof(float);  // 32 MB
    float* Are = (float*)(ws + off); off += plane;
    float* Aim = (float*)(ws + off); off += plane;
    float* Bre = (float*)(ws + off); off += plane;
    float* Bim = (float*)(ws + off);

    build_unitaries_kernel<<<12, 256, 0, stream>>>(params, Ure, Uim);

    dim3 grd(2048), blk(256);   // 2048 blocks = 128 batch x 16 column-groups
    auto UO = [](int d, int g) { return (d * 4 + g) * 256; };

    // layer 0: groups (0,1) on bits 15-8 (input real), then (2,3) in-place
    apply_pair_kernel<true,  false, true ><<<grd, blk, 0, stream>>>(
        state, nullptr, Are, Aim,
        Ure + UO(0,0), Uim + UO(0,0), Ure + UO(0,1), Uim + UO(0,1));
    apply_pair_kernel<false, false, false><<<grd, blk, 0, stream>>>(
        Are, Aim, Are, Aim,
        Ure + UO(0,2), Uim + UO(0,2), Ure + UO(0,3), Uim + UO(0,3));
    // layer 1: layer-0 CNOT ring fused into first read; A -> B, then in-place
    apply_pair_kernel<false, true,  true ><<<grd, blk, 0, stream>>>(
        Are, Aim, Bre, Bim,
        Ure + UO(1,0), Uim + UO(1,0), Ure + UO(1,1), Uim + UO(1,1));
    apply_pair_kernel<false, false, false><<<grd, blk, 0, stream>>>(
        Bre, Bim, Bre, Bim,
        Ure + UO(1,2), Uim + UO(1,2), Ure + UO(1,3), Uim + UO(1,3));
    // layer 2: layer-1 CNOT ring fused; B -> A, then in-place
    apply_pair_kernel<false, true,  true ><<<grd, blk, 0, stream>>>(
        Bre, Bim, Are, Aim,
        Ure + UO(2,0), Uim + UO(2,0), Ure + UO(2,1), Uim + UO(2,1));
    apply_pair_kernel<false, false, false><<<grd, blk, 0, stream>>>(
        Are, Aim, Are, Aim,
        Ure + UO(2,2), Uim + UO(2,2), Ure + UO(2,3), Uim + UO(2,3));
    // layer-2 CNOT ring fused (forward map) into the feature reduction
    features_kernel<<<BATCH, 256, 0, stream>>>(Are, Aim, head_w, head_b, out);
}